// SpikingGoalSelector_21294447854014
// MI455X (gfx1250) — compile-verified
//
#include <hip/hip_runtime.h>

// CDNA5 / gfx1250: wave32, WMMA (not MFMA).
typedef __attribute__((ext_vector_type(2))) float v2f;  // fp32 A/B operand of 16x16x4 (64 vals / 32 lanes)
typedef __attribute__((ext_vector_type(8))) float v8f;  // fp32 16x16 C/D accumulator

#define NGOALS 4
#define NWAVES 8

// Izhikevich 'RS' constants from the reference
#define P_A 0.02f
#define P_B 0.2f
#define P_C (-65.0f)
#define P_D 8.0f
#define I_TONIC (-3.0f)

__global__ __launch_bounds__(256) void SpikingGoalSelector_21294447854014_kernel(
    const float* __restrict__ x,        // pal_d_rate      [K]
    const float* __restrict__ bias,     // neuromod_bias   [4]
    const float* __restrict__ W,        // W_pald_goal     [4, K] row-major
    const int*   __restrict__ substeps_p,
    float* __restrict__ out,            // [6] = rates[4], winner, confidence
    int K)
{
    __shared__ float partial[NWAVES][NGOALS];

    const int tid  = threadIdx.x;
    const int wave = tid >> 5;
    const int lane = tid & 31;
    const int half = lane >> 4;   // 0 -> K pair {0,1}, 1 -> K pair {2,3}
    const int m    = lane & 15;   // A-matrix row (goal id for m<4, zero rows otherwise)

    // Hoisted multiplicative mask: rows 4..15 of A must be zero. Clamping the
    // row index to m&3 keeps every lane's load address inside W, so the inner
    // loop has NO divergence / EXEC writes around the WMMA (EXEC must be ~0).
    const float amask = (m < NGOALS) ? 1.0f : 0.0f;
    const float* __restrict__ Wrow = W + (size_t)(m & (NGOALS - 1)) * (size_t)K;

    // ---- GEMV via full-fp32 WMMA: D(16x16) += A(16x4) * B(4x16) --------------
    v8f acc = {0.f, 0.f, 0.f, 0.f, 0.f, 0.f, 0.f, 0.f};

    const int nchunk = K >> 2;                 // K consumed 4 per WMMA
    for (int i = wave; i < nchunk; i += NWAVES) {
        const int k0 = (i << 2) + (half << 1);
        // B: replicate the x chunk across all 16 columns (layout-robust in N)
        v2f b;
        b.x = x[k0];
        b.y = x[k0 + 1];
        // A: unconditional 64-bit load, zeroed by mask on rows 4..15
        v2f a;
        a.x = Wrow[k0]     * amask;
        a.y = Wrow[k0 + 1] * amask;
        // 8 args: (neg_a, A, neg_b, B, c_mod, C, reuse_a, reuse_b)
        acc = __builtin_amdgcn_wmma_f32_16x16x4_f32(false, a, false, b,
                                                    (short)0, acc, false, false);
    }

    // C/D layout: lane 0, VGPR g  ==  D[M=g][N=0]  -> per-wave partial dot for goal g
    if (lane == 0) {
        partial[wave][0] = acc[0];
        partial[wave][1] = acc[1];
        partial[wave][2] = acc[2];
        partial[wave][3] = acc[3];
    }
    __syncthreads();

    // ---- Scalar tail: normalization + 30-substep 4-neuron WTA (serial) -------
    if (tid == 0) {
        float I[NGOALS];
        float iabs = 0.f;
        for (int g = 0; g < NGOALS; ++g) {
            float s = 0.f;
            for (int w = 0; w < NWAVES; ++w) s += partial[w][g];
            I[g] = s;
            iabs += fabsf(s);
        }
        float imean = iabs * (1.0f / NGOALS) + 1e-8f;
        float scl   = (imean > 1e-3f) ? (3.0f / imean) : 0.0f;
        for (int g = 0; g < NGOALS; ++g) I[g] = I[g] * scl + 2.0f * bias[g];

        const int T = substeps_p[0];
        float v[NGOALS], u[NGOALS], rate[NGOALS], counts[NGOALS];
        for (int g = 0; g < NGOALS; ++g) {
            v[g] = P_C; u[g] = P_B * P_C; rate[g] = 0.f; counts[g] = 0.f;
        }
        for (int t = 0; t < T; ++t) {
            const float rsum = rate[0] + rate[1] + rate[2] + rate[3];
            for (int g = 0; g < NGOALS; ++g) {
                // W_rec @ rate with W_SELF=4, W_INHIB=-2:
                //   4*r[g] - 2*(rsum - r[g]) = 6*r[g] - 2*rsum
                const float Irec = 6.0f * rate[g] - 2.0f * rsum;
                const float Iin  = I[g] + Irec + I_TONIC;
                const float vo = v[g], uo = u[g];
                float vn = vo + (0.04f * vo * vo + 5.0f * vo + 140.0f - uo + Iin);
                float un = uo + P_A * (P_B * vo - uo);        // uses pre-update v
                const float sp = (vn >= 30.0f) ? 1.0f : 0.0f;
                if (sp > 0.0f) { vn = P_C; un += P_D; }
                v[g] = vn; u[g] = un;
                rate[g]   = 0.9f * rate[g] + 0.1f * sp;       // EMA drives recurrence
                counts[g] += sp;                              // counts drive output
            }
        }
        const float inv = 1.0f / (float)T;
        float rs = 0.f, best = -1.f;
        int win = 0;
        for (int g = 0; g < NGOALS; ++g) {
            const float r = counts[g] * inv;                  // rates = counts/substeps
            out[g] = r;
            rs += r;
            if (r > best) { best = r; win = g; }              // strict '>' => first argmax
        }
        out[4] = (float)win;
        out[5] = best / (rs + 1e-8f);
    }
}

extern "C" void kernel_launch(void* const* d_in, const int* in_sizes, int n_in,
                              void* d_out, int out_size, void* d_ws, size_t ws_size,
                              hipStream_t stream) {
    const float* x       = (const float*)d_in[0];   // pal_d_rate   [K]
    const float* bias    = (const float*)d_in[1];   // neuromod_bias[4]
    const float* W       = (const float*)d_in[2];   // W_pald_goal  [4, K]
    const int*   substep = (const int*)  d_in[3];   // scalar
    float*       out     = (float*)d_out;           // 6 floats
    const int K = in_sizes[0];

    SpikingGoalSelector_21294447854014_kernel<<<dim3(1), dim3(256), 0, stream>>>(
        x, bias, W, substep, out, K);
}